// BatchMultiHeadGraphAttention_79224966742096
// MI455X (gfx1250) — compile-verified
//
#include <hip/hip_runtime.h>

// ---------------------------------------------------------------------------
// BatchMultiHeadGraphAttention forward for MI455X (gfx1250, wave32, WMMA).
//   h      [8,1024,64]  f32
//   w      [8,64,32]    f32
//   a_src  [8,32,1024]  f32
//   a_dst  [8,32,1024]  f32
//   bias   [32]         f32
// Outputs (concatenated in d_out): out [8,8,1024,32], attn [8,8,1024,1024].
//
// Pipeline:
//  K0: a_sum = a_src + a_dst                                   (1 MB, L2-resident)
//  K1: h_prime = h @ w   via V_WMMA_F32_16X16X4_F32            (8 MB ws, L2-resident)
//  K2: fused per 32-row slab of one (b,head):
//      logits (WMMA) -> leaky_relu -> LDS -> softmax in LDS ->
//      attn streamed out ONCE with b128 non-temporal stores (256 MB > 192 MB L2,
//      keep the hot 9 MB working set cached) ->
//      out = attn_slab @ h_prime (WMMA, A from LDS), deterministic K-split.
// ---------------------------------------------------------------------------

typedef __attribute__((ext_vector_type(2))) float v2f;
typedef __attribute__((ext_vector_type(4))) float v4f;
typedef __attribute__((ext_vector_type(8))) float v8f;

#define BS 8
#define NH 8
#define NN 1024
#define FIN 64
#define FOUT 32
#define NEG_SLOPE 0.2f

#define ROWS 32            // rows of attn per workgroup
#define LDSS 1028          // padded LDS row stride (floats): 257*16B rows stay
                           // 16B-aligned and strided A-reads hit distinct banks

static __device__ __forceinline__ v8f wmma_f32(v2f a, v2f b, v8f c) {
  // 8-arg pattern: (neg_a, A, neg_b, B, c_mod, C, reuse_a, reuse_b)
  return __builtin_amdgcn_wmma_f32_16x16x4_f32(false, a, false, b, (short)0, c,
                                               false, false);
}

// ---------------- K0: a_sum = a_src + a_dst  (NH*FOUT*NN elements) ----------
__global__ void __launch_bounds__(256) asum_kernel(const float* __restrict__ a_src,
                                                   const float* __restrict__ a_dst,
                                                   float* __restrict__ a_sum) {
  int i = blockIdx.x * 256 + threadIdx.x;
  a_sum[i] = a_src[i] + a_dst[i];
}

// ---------------- K1: h_prime[b,hh] = h[b] @ w[hh] --------------------------
// One wave computes one 16x16 output tile; 8 waves per block.
// Tiles per (b,hh): (1024/16) * (32/16) = 128.  Total waves = 64*128 = 8192.
__global__ void __launch_bounds__(256) hprime_kernel(const float* __restrict__ h,
                                                     const float* __restrict__ w,
                                                     float* __restrict__ hp) {
  const int lane = threadIdx.x & 31;
  const int wave = threadIdx.x >> 5;
  const int gw   = blockIdx.x * 8 + wave;
  const int bh   = gw >> 7;          // (b*NH + hh)
  const int t    = gw & 127;
  const int mt   = t >> 1;           // 0..63  (row tile)
  const int nt   = t & 1;            // 0..1   (col tile)
  const int b    = bh >> 3;
  const int hh   = bh & 7;

  const int row  = lane & 15;
  const int col  = lane & 15;
  const int koff = (lane >> 4) << 1;     // 0 or 2
  const int cr   = (lane >> 4) << 3;     // 0 or 8

  const float* A = h + (size_t)b * NN * FIN;       // [1024][64]
  const float* B = w + (size_t)hh * FIN * FOUT;    // [64][32]

  v8f acc = {};
  for (int k = 0; k < FIN; k += 4) {
    v2f a, bb;
    const float* ap = A + (size_t)(mt * 16 + row) * FIN + k + koff;
    a.x = ap[0];
    a.y = ap[1];
    const float* bp = B + (size_t)(k + koff) * FOUT + nt * 16 + col;
    bb.x = bp[0];
    bb.y = bp[FOUT];
    acc = wmma_f32(a, bb, acc);
  }

  float* C = hp + ((size_t)bh * NN + mt * 16) * FOUT + nt * 16;
  for (int i = 0; i < 8; ++i)
    C[(size_t)(cr + i) * FOUT + col] = acc[i];
}

// ---------------- K2: fused logits -> softmax -> out ------------------------
// Grid: BS*NH*(NN/ROWS) = 2048 blocks of 256 threads (8 waves).
__global__ void __launch_bounds__(256) attn_kernel(const float* __restrict__ hp,
                                                   const float* __restrict__ a_sum,
                                                   const float* __restrict__ bias,
                                                   float* __restrict__ out,
                                                   float* __restrict__ attn_out) {
  __shared__ float attn_s[ROWS * LDSS];   // ~128.5 KB of the 320 KB WGP LDS

  const int lane = threadIdx.x & 31;
  const int wave = threadIdx.x >> 5;
  const int bh   = blockIdx.x >> 5;       // (b*NH + hh)
  const int rb   = blockIdx.x & 31;
  const int row0 = rb * ROWS;
  const int hh   = bh & 7;

  const int row  = lane & 15;
  const int col  = lane & 15;
  const int koff = (lane >> 4) << 1;      // 0 or 2
  const int cr   = (lane >> 4) << 3;      // 0 or 8

  const float* hp_bh = hp + (size_t)bh * NN * FOUT;          // [1024][32]
  const float* Bs    = a_sum + (size_t)hh * FOUT * NN;       // [32][1024]

  // ---- Phase 1: logits slab [32 x 1024] = hp_rows[32x32] @ a_sum[32x1024] --
  // A fragments (both 16-row tiles, all 8 k-steps) hoisted into registers.
  v2f afrag[2][8];
  for (int mt = 0; mt < 2; ++mt)
    for (int ks = 0; ks < 8; ++ks) {
      const float* ap =
          hp_bh + (size_t)(row0 + mt * 16 + row) * FOUT + ks * 4 + koff;
      afrag[mt][ks].x = ap[0];
      afrag[mt][ks].y = ap[1];
    }

  for (int jj = 0; jj < 8; ++jj) {
    const int nt = wave + 8 * jj;         // column tile 0..63
    v8f acc0 = {}, acc1 = {};
    for (int ks = 0; ks < 8; ++ks) {
      v2f b;
      const float* bp = Bs + (size_t)(ks * 4 + koff) * NN + nt * 16 + col;
      b.x = bp[0];
      b.y = bp[NN];
      acc0 = wmma_f32(afrag[0][ks], b, acc0);
      acc1 = wmma_f32(afrag[1][ks], b, acc1);
    }
    for (int i = 0; i < 8; ++i) {
      float x0 = acc0[i];
      x0 = x0 > 0.0f ? x0 : NEG_SLOPE * x0;
      attn_s[(cr + i) * LDSS + nt * 16 + col] = x0;
      float x1 = acc1[i];
      x1 = x1 > 0.0f ? x1 : NEG_SLOPE * x1;
      attn_s[(16 + cr + i) * LDSS + nt * 16 + col] = x1;
    }
  }
  __syncthreads();

  // ---- Phase 2: row softmax in LDS (float4 passes); stream attn to HBM ----
  // attn is write-once / never re-read: 256 MB > 192 MB L2, so use b128
  // non-temporal stores to avoid evicting the L2-resident h_prime/a_sum.
  for (int j = 0; j < 4; ++j) {
    const int r = wave + 8 * j;
    float* rowp = &attn_s[r * LDSS];
    float m = -3.402823466e38f;
    for (int c = lane * 4; c < NN; c += 128) {
      v4f x = *(const v4f*)&rowp[c];
      m = fmaxf(m, fmaxf(fmaxf(x.x, x.y), fmaxf(x.z, x.w)));
    }
    for (int off = 16; off > 0; off >>= 1) m = fmaxf(m, __shfl_xor(m, off));
    float s = 0.0f;
    for (int c = lane * 4; c < NN; c += 128) {
      v4f x = *(const v4f*)&rowp[c];
      x.x = __expf(x.x - m);
      x.y = __expf(x.y - m);
      x.z = __expf(x.z - m);
      x.w = __expf(x.w - m);
      *(v4f*)&rowp[c] = x;
      s += x.x + x.y + x.z + x.w;
    }
    for (int off = 16; off > 0; off >>= 1) s += __shfl_xor(s, off);
    const float inv = 1.0f / s;
    float* gout = attn_out + ((size_t)bh * NN + row0 + r) * NN;
    for (int c = lane * 4; c < NN; c += 128) {
      v4f x = *(const v4f*)&rowp[c];
      x.x *= inv; x.y *= inv; x.z *= inv; x.w *= inv;
      *(v4f*)&rowp[c] = x;                        // keep normalized slab for ph3
      __builtin_nontemporal_store(x, (v4f*)&gout[c]);  // single NT b128 stream
    }
  }
  __syncthreads();

  // ---- Phase 3: out[32x32] = attn_slab[32x1024] @ hp[1024x32] --------------
  // 4 output tiles (mt,nt), K split in halves across 8 waves; deterministic
  // pairwise combine through LDS scratch.
  const int mt = wave & 1;
  const int nt = (wave >> 1) & 1;
  const int kh = wave >> 2;               // 0 = K[0,512), 1 = K[512,1024)
  v8f acc = {};
  for (int k = kh * 512; k < kh * 512 + 512; k += 4) {
    v2f a, b;
    const float* ap = &attn_s[(mt * 16 + row) * LDSS + k + koff];
    a.x = ap[0];
    a.y = ap[1];
    const float* bp = hp_bh + (size_t)(k + koff) * FOUT + nt * 16 + col;
    b.x = bp[0];
    b.y = bp[FOUT];
    acc = wmma_f32(a, b, acc);
  }
  __syncthreads();                        // all LDS reads of attn_s done
  if (kh == 1) {
    float* sc = &attn_s[(wave - 4) * 256];
    for (int i = 0; i < 8; ++i) sc[i * 32 + lane] = acc[i];
  }
  __syncthreads();
  if (kh == 0) {
    const float* sc = &attn_s[wave * 256];
    const float bv = bias[nt * 16 + col];
    float* Cp = out + ((size_t)bh * NN + row0 + mt * 16) * FOUT + nt * 16;
    for (int i = 0; i < 8; ++i) {
      float v = acc[i] + sc[i * 32 + lane] + bv;
      Cp[(size_t)(cr + i) * FOUT + col] = v;
    }
  }
}

// ---------------------------------------------------------------------------
extern "C" void kernel_launch(void* const* d_in, const int* in_sizes, int n_in,
                              void* d_out, int out_size, void* d_ws, size_t ws_size,
                              hipStream_t stream) {
  const float* h     = (const float*)d_in[0];
  const float* w     = (const float*)d_in[1];
  const float* a_src = (const float*)d_in[2];
  const float* a_dst = (const float*)d_in[3];
  const float* bias  = (const float*)d_in[4];

  float* out      = (float*)d_out;                               // 2,097,152 f32
  float* attn     = (float*)d_out + (size_t)BS * NH * NN * FOUT; // 67,108,864 f32

  float* hp    = (float*)d_ws;                                   // 8 MB
  float* a_sum = hp + (size_t)BS * NH * NN * FOUT;               // +1 MB

  // K0: a_sum (262144 elems)
  asum_kernel<<<(NH * FOUT * NN) / 256, 256, 0, stream>>>(a_src, a_dst, a_sum);
  // K1: h_prime (8192 wave-tiles / 8 waves per block)
  hprime_kernel<<<1024, 256, 0, stream>>>(h, w, hp);
  // K2: fused attention (BS*NH*(NN/ROWS) = 2048 blocks)
  attn_kernel<<<BS * NH * (NN / ROWS), 256, 0, stream>>>(hp, a_sum, bias, out, attn);
}